// STSGNN_88278757802471
// MI455X (gfx1250) — compile-verified
//
// STSGNN forward for MI455X (gfx1250), fp32 end-to-end.
// All dense linear algebra runs through V_WMMA_F32_16X16X4_F32 (wave32 WMMA).
// Tile staging uses GLOBAL_LOAD_ASYNC_TO_LDS_B128 (ASYNCcnt-tracked DMA) with
// double-buffered LDS so the fp32 WMMA pipeline overlaps the next tile's DMA.
// Dominant cost: spatial graph mixing = per-batch 1024 x (10*1024) x 768 fp32 GEMM,
// with the 42MB sp_list operand resident in the 192MB L2 -> matrix-core bound.
//
// Workspace requirement: ~430 MB (see offsets below).

#include <hip/hip_runtime.h>
#include <math.h>

typedef float v2f __attribute__((ext_vector_type(2)));
typedef float v8f __attribute__((ext_vector_type(8)));

#define NB 16      // batch
#define NN 1024    // nodes
#define NC 64      // channels
#define NT 12      // time steps
#define SO 10      // spatial order
#define TO 5       // temporal order

// ---------------------------------------------------------------------------
// helpers
// ---------------------------------------------------------------------------
__device__ __forceinline__ float sigmoidf_(float x) { return 1.f / (1.f + expf(-x)); }
__device__ __forceinline__ float relu_(float x) { return x > 0.f ? x : 0.f; }

// lane-granular async DMA: 16 bytes global -> LDS, tracked by ASYNCcnt
__device__ __forceinline__ void async_b128(void* lds_ptr, const float* gaddr) {
  unsigned lds_off = (unsigned)(unsigned long long)lds_ptr;  // low 32 bits = LDS offset
  asm volatile("global_load_async_to_lds_b128 %0, %1, off"
               :: "v"(lds_off), "v"(gaddr)
               : "memory");
}
__device__ __forceinline__ void wait_async_le3() {
  asm volatile("s_wait_asynccnt 0x3" ::: "memory");
}
__device__ __forceinline__ void wait_async_0() {
  asm volatile("s_wait_asynccnt 0x0" ::: "memory");
}

__global__ void k_zero(float* __restrict__ p, long n) {
  long i = (long)blockIdx.x * 256 + threadIdx.x;
  if (i < n) p[i] = 0.f;
}

// ---------------------------------------------------------------------------
// tp_list: 5 weighted (I+M)^(5-i)(I-M)^i for the 12x12 temporal matrix
// ---------------------------------------------------------------------------
__global__ void k_tplist(const float* __restrict__ tp, float* __restrict__ tpl) {
  __shared__ float P1[6][144], P2[6][144], L1[144], L2[144];
  int tid = threadIdx.x;
  if (tid < 144) {
    int r = tid / 12, c = tid % 12;
    float m = tp[tid];
    float I = (r == c) ? 1.f : 0.f;
    L1[tid] = I + m; L2[tid] = I - m;
    P1[0][tid] = I;  P2[0][tid] = I;
  }
  __syncthreads();
  for (int j = 1; j <= 5; ++j) {
    if (tid < 144) {
      int r = tid / 12, c = tid % 12;
      float a1 = 0.f, a2 = 0.f;
      for (int k = 0; k < 12; ++k) {
        a1 += P1[j - 1][r * 12 + k] * L1[k * 12 + c];
        a2 += P2[j - 1][r * 12 + k] * L2[k * 12 + c];
      }
      P1[j][tid] = a1; P2[j][tid] = a2;
    }
    __syncthreads();
  }
  const float wgt[5] = {1.f/32.f, 5.f/32.f, 10.f/32.f, 10.f/32.f, 5.f/32.f};
  if (tid < 144) {
    int r = tid / 12, c = tid % 12;
    for (int i = 0; i < 5; ++i) {
      float a = 0.f;
      for (int k = 0; k < 12; ++k) a += P1[5 - i][r * 12 + k] * P2[i][k * 12 + c];
      tpl[i * 144 + tid] = wgt[i] * a;
    }
  }
}

// ---------------------------------------------------------------------------
// ST embedding: STE[b][t5][s10]
// ---------------------------------------------------------------------------
__global__ void k_ste(const float* __restrict__ src, const float* __restrict__ SE,
                      const float* __restrict__ t1w, const float* __restrict__ t1b,
                      const float* __restrict__ t2w, const float* __restrict__ t2b,
                      const float* __restrict__ s1w, const float* __restrict__ s1b,
                      const float* __restrict__ s2w, const float* __restrict__ s2b,
                      float* __restrict__ STE) {
  __shared__ float h[NB * 10 * 12];  // h[b][s10][l12]
  __shared__ float SEh[5 * 10];      // SEh[o5][l10]
  __shared__ float SEe[5 * 10];      // SEe[t5][s10]
  int tid = threadIdx.x;
  for (int i = tid; i < NB * 10 * 12; i += 256) {
    int b = i / 120, o = (i / 12) % 10, l = i % 12;
    float dowf = src[(long)b * 36864 + l * 3072 + 3 + 1];
    float todf = src[(long)b * 36864 + l * 3072 + 3 + 2];
    int d  = ((int)dowf) % 7;
    int td = ((int)todf) % 288;
    h[i] = relu_(t1w[o * 295 + d] + t1w[o * 295 + 7 + td] + t1b[o]);
  }
  if (tid < 50) {
    int o = tid / 10, l = tid % 10;
    float acc = s1b[o];
    for (int n = 0; n < NN; ++n) acc += s1w[o * NN + n] * SE[n * 10 + l];
    SEh[tid] = relu_(acc);
  }
  __syncthreads();
  if (tid < 50) {
    int t = tid / 10, s = tid % 10;
    float acc = s2b[s];
    for (int c = 0; c < 10; ++c) acc += s2w[s * 10 + c] * SEh[t * 10 + c];
    SEe[tid] = relu_(acc);
  }
  __syncthreads();
  for (int i = tid; i < NB * 5 * 10; i += 256) {
    int b = i / 50, p = (i / 10) % 5, s = i % 10;
    float acc = t2b[p];
    for (int l = 0; l < 12; ++l) acc += t2w[p * 12 + l] * h[(b * 10 + s) * 12 + l];
    STE[i] = relu_(SEe[p * 10 + s] + relu_(acc));
  }
}

// ---------------------------------------------------------------------------
// start conv: x[b,f,n,t] = start_w[f]*source[b,t,n,0] + start_b[f]
// ---------------------------------------------------------------------------
__global__ void k_start(const float* __restrict__ src, const float* __restrict__ sw,
                        const float* __restrict__ sb, float* __restrict__ x) {
  long i = (long)blockIdx.x * 256 + threadIdx.x;
  int b = (int)(i / 786432);
  int rem = (int)(i % 786432);
  int f = rem / 12288;
  int nt = rem % 12288;
  int n = nt / 12, t = nt % 12;
  x[i] = sw[f] * src[(long)b * 36864 + t * 3072 + n * 3 + 0] + sb[f];
}

// ---------------------------------------------------------------------------
// L1 = I + M, L2 = I - M, Q0 = I (1024x1024)
// ---------------------------------------------------------------------------
__global__ void k_buildL(const float* __restrict__ sp, float* __restrict__ L1,
                         float* __restrict__ L2, float* __restrict__ Q0) {
  int i = blockIdx.x * 256 + threadIdx.x;
  int r = i >> 10, c = i & 1023;
  float m = sp[i];
  float I = (r == c) ? 1.f : 0.f;
  L1[i] = I + m; L2[i] = I - m; Q0[i] = I;
}

// ---------------------------------------------------------------------------
// WMMA fragment step, shared by both GEMM kernels.
// Block tile 64(M) x 128(N), 8 waves as 2(M) x 4(N); wave tile 32x32:
// 2 A-frags x 2 B-frags -> 4 V_WMMA_F32_16X16X4_F32 per 4-K step.
// fp32 A frag 16x4: lanes 0-15 {K0,K1}, lanes 16-31 {K2,K3}, M = lane&15.
// fp32 B frag 4x16 (mirror): N = lane&15.
// A tile stride = 20 floats: rows 16B-aligned for async B128, and 20*r mod 64
// distinct for r=0..15 -> conflict-free column fragment reads.
// ---------------------------------------------------------------------------
__device__ __forceinline__ void wmma4(const float (*As)[20], const float (*Bs)[128],
                                      int wm, int wn, int kk, int half, int l16,
                                      v8f& a00, v8f& a01, v8f& a10, v8f& a11) {
  v2f af0, af1, bf0, bf1;
  af0.x = As[wm * 32 + l16][kk + 2 * half + 0];
  af0.y = As[wm * 32 + l16][kk + 2 * half + 1];
  af1.x = As[wm * 32 + 16 + l16][kk + 2 * half + 0];
  af1.y = As[wm * 32 + 16 + l16][kk + 2 * half + 1];
  bf0.x = Bs[kk + 2 * half + 0][wn * 32 + l16];
  bf0.y = Bs[kk + 2 * half + 1][wn * 32 + l16];
  bf1.x = Bs[kk + 2 * half + 0][wn * 32 + 16 + l16];
  bf1.y = Bs[kk + 2 * half + 1][wn * 32 + 16 + l16];
  a00 = __builtin_amdgcn_wmma_f32_16x16x4_f32(false, af0, false, bf0, (short)0, a00, false, false);
  a01 = __builtin_amdgcn_wmma_f32_16x16x4_f32(false, af0, false, bf1, (short)0, a01, false, false);
  a10 = __builtin_amdgcn_wmma_f32_16x16x4_f32(false, af1, false, bf0, (short)0, a10, false, false);
  a11 = __builtin_amdgcn_wmma_f32_16x16x4_f32(false, af1, false, bf1, (short)0, a11, false, false);
}

// issue one 64x16 A tile + 16x128 B tile via 3 async B128s per thread
__device__ __forceinline__ void stage_issue(const float* A, int lda,
                                            const float* B, int ldb,
                                            int row_base, int col_base, int k0,
                                            float (*As)[20], float (*Bs)[128],
                                            int ar, int ac, int brow, int bc4) {
  async_b128(&As[ar][ac],        &A[(long)(row_base + ar) * lda + k0 + ac]);
  async_b128(&Bs[brow][bc4],     &B[(long)(k0 + brow) * ldb + col_base + bc4]);
  async_b128(&Bs[brow + 8][bc4], &B[(long)(k0 + brow + 8) * ldb + col_base + bc4]);
}

// ---------------------------------------------------------------------------
// Generic fp32 WMMA GEMM: C[M,N] = alpha*A[M,K]@B[K,N] (+C if accumulate)
// Row-major. M mult of 64, N mult of 128, K mult of 16. 256 threads.
// ---------------------------------------------------------------------------
__launch_bounds__(256)
__global__ void k_gemm(const float* __restrict__ A, int lda,
                       const float* __restrict__ B, int ldb,
                       float* __restrict__ C, int ldc,
                       int K, float alpha, int accumulate) {
  __shared__ float As[2][64][20];
  __shared__ float Bs[2][16][128];
  const int tid = threadIdx.x;
  const int wid = tid >> 5, lane = tid & 31;
  const int wm = wid >> 2, wn = wid & 3;
  const int row_base = blockIdx.y * 64;
  const int col_base = blockIdx.x * 128;
  const int ar = tid >> 2, ac = (tid & 3) << 2;     // A stage: 64x16
  const int brow = tid >> 5, bc4 = lane << 2;       // B stage: 16x128, 2 rows/thread
  const int half = lane >> 4, l16 = lane & 15;

  v8f a00 = {0.f,0.f,0.f,0.f,0.f,0.f,0.f,0.f};
  v8f a01 = a00, a10 = a00, a11 = a00;

  const int ntiles = K >> 4;
  stage_issue(A, lda, B, ldb, row_base, col_base, 0, As[0], Bs[0], ar, ac, brow, bc4);
  for (int i = 0; i < ntiles; ++i) {
    const int buf = i & 1;
    const bool more = (i + 1 < ntiles);
    if (more) {
      stage_issue(A, lda, B, ldb, row_base, col_base, (i + 1) << 4,
                  As[buf ^ 1], Bs[buf ^ 1], ar, ac, brow, bc4);
      wait_async_le3();   // current tile's 3 DMAs done (in-order), next 3 in flight
    } else {
      wait_async_0();
    }
    __syncthreads();
#pragma unroll
    for (int kk = 0; kk < 16; kk += 4)
      wmma4(As[buf], Bs[buf], wm, wn, kk, half, l16, a00, a01, a10, a11);
    __syncthreads();      // all reads of buf done before it is refilled
  }
  // C/D layout: VGPR r -> M = r (lanes 0-15) / 8+r (lanes 16-31); N = lane&15
#pragma unroll
  for (int r = 0; r < 8; ++r) {
    int r0 = row_base + wm * 32 + half * 8 + r;
    int r1 = r0 + 16;
    int c0 = col_base + wn * 32 + l16;
    int c1 = c0 + 16;
    float v00 = alpha * a00[r], v01 = alpha * a01[r];
    float v10 = alpha * a10[r], v11 = alpha * a11[r];
    if (accumulate) {
      v00 += C[(long)r0 * ldc + c0]; v01 += C[(long)r0 * ldc + c1];
      v10 += C[(long)r1 * ldc + c0]; v11 += C[(long)r1 * ldc + c1];
    }
    C[(long)r0 * ldc + c0] = v00; C[(long)r0 * ldc + c1] = v01;
    C[(long)r1 * ldc + c0] = v10; C[(long)r1 * ldc + c1] = v11;
  }
}

// ---------------------------------------------------------------------------
// Temporal mix: Xb[o][n][f*12+t] = sum_k tweight[b,o,t,k] * x[b,f,n,k]
// ---------------------------------------------------------------------------
__global__ void k_xt(const float* __restrict__ x, const float* __restrict__ tw,
                     int b, float* __restrict__ Xb) {
  __shared__ float tws[144];
  int gid = blockIdx.x * 256 + threadIdx.x;
  int o = gid >> 16;          // 65536 per o; block never spans two o's
  int idx2 = gid & 65535;
  int f = idx2 >> 10;
  int n = idx2 & 1023;
  if (threadIdx.x < 144) tws[threadIdx.x] = tw[b * 1440 + o * 144 + threadIdx.x];
  __syncthreads();
  const float* xr = x + (long)b * 786432 + f * 12288 + n * 12;
  float xv[12];
#pragma unroll
  for (int k = 0; k < 12; ++k) xv[k] = xr[k];
  float* orow = Xb + (long)o * 786432 + n * 768 + f * 12;
#pragma unroll
  for (int t = 0; t < 12; ++t) {
    float acc = 0.f;
#pragma unroll
    for (int k = 0; k < 12; ++k) acc += tws[t * 12 + k] * xv[k];
    orow[t] = acc;
  }
}

// ---------------------------------------------------------------------------
// Spatial graph mix: for one batch b,
//   xst[b,f,m,t] = sum_o sp[o][m,:] @ Xb[o][:, f*12+t]
// M=1024, N=768, K=10*1024, register-accumulated across the 10 orders.
// ---------------------------------------------------------------------------
__device__ __forceinline__ void spat_issue(const float* SP, const float* Xb,
                                           int row_base, int col_base, int i,
                                           float (*As)[20], float (*Bs)[128],
                                           int ar, int ac, int brow, int bc4) {
  int o = i >> 6;             // 64 tiles of 16 per order
  int k0 = (i & 63) << 4;
  const float* A = SP + (long)o * 1048576;
  const float* B = Xb + (long)o * 786432;
  async_b128(&As[ar][ac],        &A[(long)(row_base + ar) * 1024 + k0 + ac]);
  async_b128(&Bs[brow][bc4],     &B[(long)(k0 + brow) * 768 + col_base + bc4]);
  async_b128(&Bs[brow + 8][bc4], &B[(long)(k0 + brow + 8) * 768 + col_base + bc4]);
}

__launch_bounds__(256)
__global__ void k_spatial(const float* __restrict__ SP, const float* __restrict__ Xb,
                          float* __restrict__ xst_b) {
  __shared__ float As[2][64][20];
  __shared__ float Bs[2][16][128];
  const int tid = threadIdx.x;
  const int wid = tid >> 5, lane = tid & 31;
  const int wm = wid >> 2, wn = wid & 3;
  const int row_base = blockIdx.y * 64;
  const int col_base = blockIdx.x * 128;
  const int ar = tid >> 2, ac = (tid & 3) << 2;
  const int brow = tid >> 5, bc4 = lane << 2;
  const int half = lane >> 4, l16 = lane & 15;

  v8f a00 = {0.f,0.f,0.f,0.f,0.f,0.f,0.f,0.f};
  v8f a01 = a00, a10 = a00, a11 = a00;

  const int ntiles = SO * 64;   // 640 K-tiles across the 10 orders
  spat_issue(SP, Xb, row_base, col_base, 0, As[0], Bs[0], ar, ac, brow, bc4);
  for (int i = 0; i < ntiles; ++i) {
    const int buf = i & 1;
    const bool more = (i + 1 < ntiles);
    if (more) {
      spat_issue(SP, Xb, row_base, col_base, i + 1,
                 As[buf ^ 1], Bs[buf ^ 1], ar, ac, brow, bc4);
      wait_async_le3();
    } else {
      wait_async_0();
    }
    __syncthreads();
#pragma unroll
    for (int kk = 0; kk < 16; kk += 4)
      wmma4(As[buf], Bs[buf], wm, wn, kk, half, l16, a00, a01, a10, a11);
    __syncthreads();
  }
  // scatter: col j -> (f = j/12, t = j%12); xst layout (f, m, t)
#pragma unroll
  for (int r = 0; r < 8; ++r) {
    int r0 = row_base + wm * 32 + half * 8 + r;
    int r1 = r0 + 16;
    int c0 = col_base + wn * 32 + l16;
    int c1 = c0 + 16;
    xst_b[(long)(c0 / 12) * 12288 + r0 * 12 + (c0 % 12)] = a00[r];
    xst_b[(long)(c1 / 12) * 12288 + r0 * 12 + (c1 % 12)] = a01[r];
    xst_b[(long)(c0 / 12) * 12288 + r1 * 12 + (c0 % 12)] = a10[r];
    xst_b[(long)(c1 / 12) * 12288 + r1 * 12 + (c1 % 12)] = a11[r];
  }
}

// ---------------------------------------------------------------------------
// theta + tweight per layer (one block per batch)
// ---------------------------------------------------------------------------
__global__ void k_theta(const float* __restrict__ STE,
                        const float* __restrict__ th1w, const float* __restrict__ th1b,
                        const float* __restrict__ th2w, const float* __restrict__ th2b,
                        const float* __restrict__ tpl, float* __restrict__ tw) {
  int b = blockIdx.x, tid = threadIdx.x;
  __shared__ float a[50], th[50];
  if (tid < 50) {
    int p = tid / 10, s = tid % 10;
    float acc = th1b[p];
    for (int t = 0; t < 5; ++t) acc += th1w[p * 5 + t] * STE[(b * 5 + t) * 10 + s];
    a[tid] = acc;
  }
  __syncthreads();
  if (tid < 50) {
    int t = tid / 10, p = tid % 10;
    float acc = th2b[p];
    for (int s = 0; s < 10; ++s) acc += th2w[p * 10 + s] * a[t * 10 + s];
    th[t * 10 + p] = relu_(acc);
  }
  __syncthreads();
  for (int i = tid; i < 10 * 144; i += 256) {
    int s = i / 144, nm = i % 144;
    float acc = 0.f;
    for (int t = 0; t < 5; ++t) acc += th[t * 10 + s] * tpl[t * 144 + nm];
    tw[b * 1440 + i] = acc;
  }
}

// ---------------------------------------------------------------------------
// h = bn(st_mlp GEMM out + bias) epilogue (elementwise)
// ---------------------------------------------------------------------------
__global__ void k_hbn(float* __restrict__ h, const float* __restrict__ mb,
                      const float* __restrict__ g, const float* __restrict__ bb) {
  long i = (long)blockIdx.x * 256 + threadIdx.x;
  int f = (int)((i / 12288) & 63);
  float s = g[f] * rsqrtf(1.f + 1e-5f);
  h[i] = h[i] * s + mb[f] * s + bb[f];
}

// ---------------------------------------------------------------------------
// Pyramid GLUs: x1 (k=1), x2 (k=3,s=3 -> T=4), x3 (k=6,s=6 -> T=2)
// ---------------------------------------------------------------------------
__global__ void k_pyr(const float* __restrict__ h,
                      const float* __restrict__ w1, const float* __restrict__ b1,
                      const float* __restrict__ w2, const float* __restrict__ b2,
                      const float* __restrict__ w3, const float* __restrict__ b3,
                      float* __restrict__ x1, float* __restrict__ x2, float* __restrict__ x3) {
  __shared__ float hs[64][12];
  int bn = blockIdx.x, b = bn >> 10, n = bn & 1023, tid = threadIdx.x;
  const float* hb = h + (long)b * 786432 + n * 12;
  for (int i = tid; i < 768; i += 256) hs[i / 12][i % 12] = hb[(long)(i / 12) * 12288 + (i % 12)];
  __syncthreads();
  for (int i = tid; i < 768; i += 256) {
    int f = i / 12, t = i % 12;
    float s = b1[f], g = b1[f + 64];
    for (int c = 0; c < 64; ++c) {
      float hv = hs[c][t];
      s += w1[f * 64 + c] * hv;
      g += w1[(f + 64) * 64 + c] * hv;
    }
    x1[(long)b * 786432 + f * 12288 + n * 12 + t] = sigmoidf_(s) * tanhf(g);
  }
  {
    int f = tid >> 2, tp = tid & 3;
    float s = b2[f], g = b2[f + 64];
    for (int c = 0; c < 64; ++c)
      for (int k = 0; k < 3; ++k) {
        float hv = hs[c][3 * tp + k];
        s += w2[(f * 64 + c) * 3 + k] * hv;
        g += w2[((f + 64) * 64 + c) * 3 + k] * hv;
      }
    x2[(long)b * 262144 + f * 4096 + n * 4 + tp] = sigmoidf_(s) * tanhf(g);
  }
  if (tid < 128) {
    int f = tid >> 1, tp = tid & 1;
    float s = b3[f], g = b3[f + 64];
    for (int c = 0; c < 64; ++c)
      for (int k = 0; k < 6; ++k) {
        float hv = hs[c][6 * tp + k];
        s += w3[(f * 64 + c) * 6 + k] * hv;
        g += w3[((f + 64) * 64 + c) * 6 + k] * hv;
      }
    x3[(long)b * 131072 + f * 2048 + n * 2 + tp] = sigmoidf_(s) * tanhf(g);
  }
}

// ---------------------------------------------------------------------------
// resize(linear, half-pixel, edge clamp) + concat + pyr_conv + bn
// + skip update + residual bn (writes new x). One block per (b,n).
// ---------------------------------------------------------------------------
__global__ void k_fuse(const float* __restrict__ x1, const float* __restrict__ x2,
                       const float* __restrict__ x3,
                       const float* __restrict__ pcw, const float* __restrict__ pcb,
                       const float* __restrict__ pg,  const float* __restrict__ pbb,
                       const float* __restrict__ skw, const float* __restrict__ skb,
                       const float* __restrict__ bg,  const float* __restrict__ bb2,
                       float* __restrict__ x, float* __restrict__ skip) {
  __shared__ float cat[192][12];
  __shared__ float fus[64][12];
  __shared__ float x2s[64][4];
  __shared__ float x3s[64][2];
  int bn = blockIdx.x, b = bn >> 10, n = bn & 1023, tid = threadIdx.x;
  for (int i = tid; i < 768; i += 256)
    cat[i / 12][i % 12] = x1[(long)b * 786432 + (i / 12) * 12288 + n * 12 + (i % 12)];
  {
    int i = tid;
    x2s[i / 4][i % 4] = x2[(long)b * 262144 + (i / 4) * 4096 + n * 4 + (i % 4)];
  }
  if (tid < 128) x3s[tid / 2][tid % 2] = x3[(long)b * 131072 + (tid / 2) * 2048 + n * 2 + (tid % 2)];
  __syncthreads();
  for (int i = tid; i < 768; i += 256) {
    int c = i / 12, t = i % 12;
    float coord = (t + 0.5f) / 3.0f - 0.5f;  // 4 -> 12
    float v;
    if (coord <= 0.f) v = x2s[c][0];
    else if (coord >= 3.f) v = x2s[c][3];
    else { int i0 = (int)coord; float fr = coord - i0; v = x2s[c][i0] * (1.f - fr) + x2s[c][i0 + 1] * fr; }
    cat[64 + c][t] = v;
    float c3 = (t + 0.5f) / 6.0f - 0.5f;     // 2 -> 12
    float v3;
    if (c3 <= 0.f) v3 = x3s[c][0];
    else if (c3 >= 1.f) v3 = x3s[c][1];
    else v3 = x3s[c][0] * (1.f - c3) + x3s[c][1] * c3;
    cat[128 + c][t] = v3;
  }
  __syncthreads();
  float inv = rsqrtf(1.f + 1e-5f);
  for (int i = tid; i < 768; i += 256) {
    int f = i / 12, t = i % 12;
    float acc = 0.f;
    for (int c = 0; c < 192; ++c) acc += pcw[f * 192 + c] * cat[c][t];
    float s1 = pg[f] * inv;
    float fv = acc * s1 + pcb[f] * s1 + pbb[f];
    fus[f][t] = fv;
    long xi = (long)b * 786432 + f * 12288 + n * 12 + t;
    float res = x[xi];
    x[xi] = (fv + res) * (bg[f] * inv) + bb2[f];
  }
  __syncthreads();
  for (int i = tid; i < 1536; i += 256) {
    int f2 = i / 12, t = i % 12;
    float acc = skb[f2];
    for (int f = 0; f < 64; ++f) acc += skw[f2 * 64 + f] * fus[f][t];
    long si = (long)b * 1572864 + f2 * 12288 + n * 12 + t;
    skip[si] += acc;
  }
}

// ---------------------------------------------------------------------------
// head: relu(skip[..., -1]) -> end1 -> end2 ; out (B,12,N,1)
// ---------------------------------------------------------------------------
__global__ void k_final(const float* __restrict__ skip,
                        const float* __restrict__ e1w, const float* __restrict__ e1b,
                        const float* __restrict__ e2w, const float* __restrict__ e2b,
                        float* __restrict__ out) {
  __shared__ float sl[128];
  __shared__ float e1[256];
  int bn = blockIdx.x, b = bn >> 10, n = bn & 1023, tid = threadIdx.x;
  if (tid < 128) sl[tid] = relu_(skip[(long)b * 1572864 + tid * 12288 + n * 12 + 11]);
  __syncthreads();
  {
    float acc = e1b[tid];
    for (int f = 0; f < 128; ++f) acc += e1w[tid * 128 + f] * sl[f];
    e1[tid] = relu_(acc);
  }
  __syncthreads();
  if (tid < 12) {
    float acc = e2b[tid];
    for (int e = 0; e < 256; ++e) acc += e2w[tid * 256 + e] * e1[e];
    out[(long)b * 12288 + tid * 1024 + n] = acc;
  }
}

// ---------------------------------------------------------------------------
// host side
// ---------------------------------------------------------------------------
struct LayerPtrs {  // jax sorted-key order within each layer dict
  const float *bn_b, *bn_g, *pyr1_b, *pyr1_w, *pyr2_b, *pyr2_w, *pyr3_b, *pyr3_w,
              *pyr_bn_b, *pyr_bn_g, *pyr_conv_b, *pyr_conv_w, *skip_b, *skip_w,
              *st_bn_b, *st_bn_g, *st_mlp_b, *st_mlp_w,
              *theta1_b, *theta1_w, *theta2_b, *theta2_w;
};

extern "C" void kernel_launch(void* const* d_in, const int* in_sizes, int n_in,
                              void* d_out, int out_size, void* d_ws, size_t ws_size,
                              hipStream_t stream) {
  (void)in_sizes; (void)n_in; (void)out_size; (void)ws_size;
  // Input order: setup_inputs dict order, params flattened as a jax pytree
  // (dict keys sorted alphabetically, 'layers' list in order):
  //  0 source  1 targets  2 sp_matrix  3 tp_matrix
  //  4 SE  5 end1_b  6 end1_w  7 end2_b  8 end2_w
  //  9..30 layer0 (22 sorted entries)  31..52 layer1
  //  53 smlp1_b 54 smlp1_w 55 smlp2_b 56 smlp2_w 57 start_b 58 start_w
  //  59 tmlp1_b 60 tmlp1_w 61 tmlp2_b 62 tmlp2_w
  const float* src    = (const float*)d_in[0];
  const float* spmat  = (const float*)d_in[2];
  const float* tpmat  = (const float*)d_in[3];
  const float* SE     = (const float*)d_in[4];
  const float* end1_b = (const float*)d_in[5];
  const float* end1_w = (const float*)d_in[6];
  const float* end2_b = (const float*)d_in[7];
  const float* end2_w = (const float*)d_in[8];
  LayerPtrs lp[2];
  for (int l = 0; l < 2; ++l) {
    const float** q = (const float**)&lp[l];
    for (int k = 0; k < 22; ++k) q[k] = (const float*)d_in[9 + 22 * l + k];
  }
  const float* smlp1_b = (const float*)d_in[53];
  const float* smlp1_w = (const float*)d_in[54];
  const float* smlp2_b = (const float*)d_in[55];
  const float* smlp2_w = (const float*)d_in[56];
  const float* start_b = (const float*)d_in[57];
  const float* start_w = (const float*)d_in[58];
  const float* tmlp1_b = (const float*)d_in[59];
  const float* tmlp1_w = (const float*)d_in[60];
  const float* tmlp2_b = (const float*)d_in[61];
  const float* tmlp2_w = (const float*)d_in[62];

  // workspace layout (floats); total ~107M floats (~430 MB)
  float* ws = (float*)d_ws;
  const long M2 = 1048576;         // 1024*1024
  float* tpl   = ws + 0;           // 720
  float* STE   = ws + 1024;        // 800
  float* twb   = ws + 2048;        // 23040
  float* L1    = ws + 32768;       // 1M
  float* L2    = L1 + M2;
  float* P0    = L2 + M2;
  float* P1    = P0 + M2;
  float* Qarr  = P1 + M2;          // 10M ; reused as per-batch Xb (7.86M) later
  float* SPL   = Qarr + 10 * M2;   // 10M
  float* xbuf  = SPL + 10 * M2;    // 12,582,912  (B,C,N,T)
  float* xst   = xbuf + 12582912;  // 12,582,912
  float* hbuf  = xst + 12582912;   // 12,582,912
  float* x1b   = hbuf + 12582912;  // 12,582,912
  float* x2b   = x1b + 12582912;   // 4,194,304
  float* x3b   = x2b + 4194304;    // 2,097,152
  float* skipb = x3b + 2097152;    // 25,165,824

  dim3 blk(256);
  auto gemm = [&](const float* A, int lda, const float* B, int ldb,
                  float* C, int ldc, int M, int Nn, int K, float alpha, int acc) {
    dim3 grid(Nn / 128, M / 64);
    k_gemm<<<grid, blk, 0, stream>>>(A, lda, B, ldb, C, ldc, K, alpha, acc);
  };

  // init
  k_zero<<<(25165824 + 255) / 256, blk, 0, stream>>>(skipb, 25165824L);
  k_tplist<<<1, blk, 0, stream>>>(tpmat, tpl);
  k_ste<<<1, blk, 0, stream>>>(src, SE, tmlp1_w, tmlp1_b, tmlp2_w, tmlp2_b,
                               smlp1_w, smlp1_b, smlp2_w, smlp2_b, STE);
  k_start<<<49152, blk, 0, stream>>>(src, start_w, start_b, xbuf);
  k_buildL<<<4096, blk, 0, stream>>>(spmat, L1, L2, Qarr);

  // sp_list[i] = C(10,i)/1024 * (I+M)^(10-i) @ (I-M)^i   (28 WMMA GEMMs, 1024^3)
  const double C10[10] = {1, 10, 45, 120, 210, 252, 210, 120, 45, 10};
  for (int i = 1; i <= 9; ++i)
    gemm(Qarr + (long)(i - 1) * M2, 1024, L2, 1024, Qarr + (long)i * M2, 1024,
         1024, 1024, 1024, 1.f, 0);
  const float* Pcur = L1;
  float* Pping[2] = {P0, P1};
  int pb = 0;
  for (int e = 1; e <= 10; ++e) {
    int i = 10 - e;
    float w = (float)(C10[i] / 1024.0);
    gemm(Pcur, 1024, Qarr + (long)i * M2, 1024, SPL + (long)i * M2, 1024,
         1024, 1024, 1024, w, 0);
    if (e < 10) {
      gemm(Pcur, 1024, L1, 1024, Pping[pb], 1024, 1024, 1024, 1024, 1.f, 0);
      Pcur = Pping[pb]; pb ^= 1;
    }
  }

  float* Xb = Qarr;  // reuse Q region as per-batch temporal-mix scratch

  for (int l = 0; l < 2; ++l) {
    const LayerPtrs& L = lp[l];
    k_theta<<<NB, blk, 0, stream>>>(STE, L.theta1_w, L.theta1_b,
                                    L.theta2_w, L.theta2_b, tpl, twb);
    for (int b = 0; b < NB; ++b) {
      k_xt<<<2560, blk, 0, stream>>>(xbuf, twb, b, Xb);
      dim3 gsp(6, 16);  // N=768 cols, M=1024 rows
      k_spatial<<<gsp, blk, 0, stream>>>(SPL, Xb, xst + (long)b * 786432);
    }
    // st_mlp: h = W[:, :64]@x + W[:, 64:]@xst  (per-batch WMMA GEMMs, M=64,K=64,N=12288)
    for (int b = 0; b < NB; ++b) {
      gemm(L.st_mlp_w,      128, xbuf + (long)b * 786432, 12288,
           hbuf + (long)b * 786432, 12288, 64, 12288, 64, 1.f, 0);
      gemm(L.st_mlp_w + 64, 128, xst  + (long)b * 786432, 12288,
           hbuf + (long)b * 786432, 12288, 64, 12288, 64, 1.f, 1);
    }
    k_hbn<<<49152, blk, 0, stream>>>(hbuf, L.st_mlp_b, L.st_bn_g, L.st_bn_b);
    k_pyr<<<NB * NN, blk, 0, stream>>>(hbuf, L.pyr1_w, L.pyr1_b, L.pyr2_w, L.pyr2_b,
                                       L.pyr3_w, L.pyr3_b, x1b, x2b, x3b);
    k_fuse<<<NB * NN, blk, 0, stream>>>(x1b, x2b, x3b,
                                        L.pyr_conv_w, L.pyr_conv_b, L.pyr_bn_g, L.pyr_bn_b,
                                        L.skip_w, L.skip_b, L.bn_g, L.bn_b,
                                        xbuf, skipb);
  }
  k_final<<<NB * NN, blk, 0, stream>>>(skipb, end1_w, end1_b, end2_w, end2_b,
                                       (float*)d_out);
}